// BurstDemosaicking_34419867910154
// MI455X (gfx1250) — compile-verified
//
#include <hip/hip_runtime.h>
#include <stdint.h>

namespace {
constexpr int Nf = 8, H = 1536, W = 1536;
constexpr int HWc = H * W;                        // 2,359,296
constexpr int BLK = 256;                          // 8 waves (wave32)
constexpr int PPT = 8;                            // pixels per thread (2 x b128 async)
constexpr int TILE = BLK * PPT;                   // 2048
constexpr int TILES_PER_FRAME = HWc / TILE;       // 1152 (exact)
constexpr int G_RES = Nf * TILES_PER_FRAME;       // 9216 residual blocks
constexpr int PPT_P = 4;
constexpr int TILE_P = BLK * PPT_P;               // 1024
constexpr int ELEMS = 3 * HWc;                    // 7,077,888
constexpr int G_PRI = ELEMS / TILE_P;             // 6912 prior blocks (exact)
constexpr int G_ALL = G_RES + G_PRI;              // 16128 blocks, one fused grid
}

// 16-byte GCC-style vector type matching the async-LDS builtin's pointee
typedef int v4i_b128 __attribute__((vector_size(16)));

__device__ __forceinline__ float warp_reduce(float v) {
#pragma unroll
  for (int o = 16; o > 0; o >>= 1) v += __shfl_xor(v, o, 32);
  return v;
}

// Valid in thread 0 after return. Internally synchronized (safe to call twice).
__device__ __forceinline__ float block_reduce(float v, float* red) {
  v = warp_reduce(v);
  const int lane = threadIdx.x & 31, wv = threadIdx.x >> 5;
  if (lane == 0) red[wv] = v;
  __syncthreads();
  float r = (threadIdx.x < (BLK >> 5)) ? red[threadIdx.x] : 0.0f;
  if (wv == 0) r = warp_reduce(r);
  __syncthreads();
  return r;
}

__device__ __forceinline__ void async_copy_b128_to_lds(const float* gsrc, float* lds_dst) {
#if defined(__HIP_DEVICE_COMPILE__) && __has_builtin(__builtin_amdgcn_global_load_async_to_lds_b128)
  __builtin_amdgcn_global_load_async_to_lds_b128(
      (__attribute__((address_space(1))) v4i_b128*)gsrc,
      (__attribute__((address_space(3))) v4i_b128*)lds_dst,
      0, 0);
#else
  unsigned lds_off = (unsigned)(uintptr_t)lds_dst;  // low 32 bits = LDS byte address
  asm volatile("global_load_async_to_lds_b128 %0, %1, off"
               :: "v"(lds_off), "v"((unsigned long long)(uintptr_t)gsrc)
               : "memory");
#endif
}

__device__ __forceinline__ void wait_asynccnt0() {
#if defined(__HIP_DEVICE_COMPILE__) && __has_builtin(__builtin_amdgcn_s_wait_asynccnt)
  __builtin_amdgcn_s_wait_asynccnt(0);
#else
  asm volatile("s_wait_asynccnt 0x0" ::: "memory");
#endif
  asm volatile("" ::: "memory");  // keep LDS reads below the wait
}

// ~1-ulp reciprocal: hardware v_rcp_f32 + one Newton-Raphson step (3 VALU ops)
__device__ __forceinline__ float fast_rcp(float x) {
#if defined(__HIP_DEVICE_COMPILE__)
  float r = __builtin_amdgcn_rcpf(x);
  return fmaf(fmaf(-x, r, 1.0f), r, r);
#else
  return 1.0f / x;
#endif
}

// One fused grid: blocks [0, G_RES) compute reprojection-residual partials,
// blocks [G_RES, G_ALL) compute gradient-prior partials. Disjoint work, same
// partial-sum protocol -> prior blocks backfill WGPs during the residual tail.
__global__ void __launch_bounds__(BLK) fused_kernel(
    const float* __restrict__ inputs, const float* __restrict__ hom,
    const float* __restrict__ rec, float* __restrict__ partials) {
  __shared__ float tile[TILE];
  __shared__ float red[BLK / 32];
  const int tid = threadIdx.x;
  const int bid = blockIdx.x;

  if (bid < G_RES) {
    // ---------------- reprojection residuals ----------------
    const int base = bid * TILE;  // < 18.9M, fits int

    // CDNA5 async path: stream this block's `inputs` slice into LDS (32B/thread)
    async_copy_b128_to_lds(inputs + base + tid * PPT,     &tile[tid * PPT]);
    async_copy_b128_to_lds(inputs + base + tid * PPT + 4, &tile[tid * PPT + 4]);

    // Per-frame homography (uniform across block -> scalar loads)
    const int n = bid / TILES_PER_FRAME;
    const float* hp = hom + n * 8;
    const float h0 = hp[0], h1 = hp[1], h2 = hp[2], h3 = hp[3];
    const float h4 = hp[4], h5 = hp[5], h6 = hp[6], h7 = hp[7];

    // Frame-local flat pixel index of this thread's first pixel.
    // PPT | W, so all PPT pixels share one row: single divide per thread.
    // A wave spans 256 consecutive pixels and 256 | W, so y is WAVE-UNIFORM:
    // force it scalar so the Bayer plane bases live in SGPRs (GVS-form gathers).
    const int p0 = (base - n * HWc) + tid * PPT;
    const int y  = __builtin_amdgcn_readfirstlane(p0 / W);
    const int x0 = p0 - y * W;           // per-lane, even (multiple of 8)
    const float yf = (float)y, xf0 = (float)x0;

    // Homography numerators/denominator at (x0, y); advance by +h0/+h3/+h6 per x.
    float nx = fmaf(h0, xf0, fmaf(h1, yf, h2));
    float ny = fmaf(h3, xf0, fmaf(h4, yf, h5));
    float dn = fmaf(h6, xf0, fmaf(h7, yf, 1.0f));

    // Bayer planes (scalar): x parity alternates starting even; y parity fixed.
    const int ye = y & 1;
    const float* plane_ev = rec + (ye ? HWc : 0);        // c = ye?1:0
    const float* plane_od = rec + (ye ? 2 * HWc : HWc);  // c = ye?2:1

    float s[PPT];
    bool vld[PPT];
#pragma unroll
    for (int j = 0; j < PPT; ++j) {
      const float inv = fast_rcp(dn);
      const float xw = nx * inv;
      const float yw = ny * inv;
      vld[j] = (xw >= 0.0f) & (xw <= (float)(W - 1)) &
               (yw >= 0.0f) & (yw <= (float)(H - 1));
      const float fx0 = floorf(xw), fy0 = floorf(yw);
      const float fx = xw - fx0, fy = yw - fy0;
      const int x0i = (int)fx0, y0i = (int)fy0;
      const int xc0 = min(max(x0i, 0), W - 1);
      const int xc1 = min(max(x0i + 1, 0), W - 1);
      const int yc0 = min(max(y0i, 0), H - 1);
      const int yc1 = min(max(y0i + 1, 0), H - 1);
      const float* pl = (j & 1) ? plane_od : plane_ev;   // scalar base
      const int row0 = yc0 * W, row1 = yc1 * W;          // 32-bit indices
      const float p00 = pl[row0 + xc0], p01 = pl[row0 + xc1];
      const float p10 = pl[row1 + xc0], p11 = pl[row1 + xc1];
      const float top = fmaf(fx, p01 - p00, p00);
      const float bot = fmaf(fx, p11 - p10, p10);
      s[j] = fmaf(fy, bot - top, top);
      nx += h0; ny += h3; dn += h6;
    }

    wait_asynccnt0();
    const float4 inA = *reinterpret_cast<const float4*>(&tile[tid * PPT]);
    const float4 inB = *reinterpret_cast<const float4*>(&tile[tid * PPT + 4]);
    const float inj[PPT] = {inA.x, inA.y, inA.z, inA.w, inB.x, inB.y, inB.z, inB.w};

    float acc = 0.0f;
#pragma unroll
    for (int j = 0; j < PPT; ++j) {
      // Select (not multiply): invalid lanes must contribute exactly 0 even if
      // s[j] is inf/NaN, matching jnp.where(valid, s - inputs, 0).
      const float r = vld[j] ? (s[j] - inj[j]) : 0.0f;
      acc = fmaf(r, r, acc);
    }

    const float tot = block_reduce(acc, red);
    if (tid == 0) partials[bid] = tot;
  } else {
    // ---------------- gradient prior ----------------
    const int base = (bid - G_RES) * TILE_P + tid * PPT_P;  // < 7.1M, fits int
    const int row = base / W;          // 0 .. 3H-1 (W%4==0 so groups stay in-row)
    const int x = base - row * W;
    const int y = row % H;

    const float* p = rec + base;
    const float4 a = *reinterpret_cast<const float4*>(p);
    float acc = 0.0f, d;
    d = a.y - a.x; acc = fmaf(d, d, acc);
    d = a.z - a.y; acc = fmaf(d, d, acc);
    d = a.w - a.z; acc = fmaf(d, d, acc);
    if (x + 4 < W) { const float e = p[4]; d = e - a.w; acc = fmaf(d, d, acc); }
    if (y < H - 1) {
      const float4 b = *reinterpret_cast<const float4*>(p + W);
      d = b.x - a.x; acc = fmaf(d, d, acc);
      d = b.y - a.y; acc = fmaf(d, d, acc);
      d = b.z - a.z; acc = fmaf(d, d, acc);
      d = b.w - a.w; acc = fmaf(d, d, acc);
    }

    const float tot = block_reduce(acc, red);
    if (tid == 0) partials[bid] = tot;
  }
}

__global__ void __launch_bounds__(BLK) finalize_kernel(
    const float* __restrict__ pr, int nr, const float* __restrict__ pp, int np,
    const float* __restrict__ gw, float* __restrict__ out) {
  __shared__ float red[BLK / 32];
  float a = 0.0f, b = 0.0f;
  for (int i = threadIdx.x; i < nr; i += BLK) a += pr[i];
  for (int i = threadIdx.x; i < np; i += BLK) b += pp[i];
  const float A = block_reduce(a, red);
  const float B = block_reduce(b, red);
  if (threadIdx.x == 0) {
    out[0] = fmaf(gw[0], B, A);  // total objective
    out[1] = A;                  // reprojection error
  }
}

extern "C" void kernel_launch(void* const* d_in, const int* in_sizes, int n_in,
                              void* d_out, int out_size, void* d_ws, size_t ws_size,
                              hipStream_t stream) {
  const float* inputs = (const float*)d_in[0];  // (8,1536,1536) f32
  const float* hom    = (const float*)d_in[1];  // (8,8) f32
  const float* rec    = (const float*)d_in[2];  // (3,1536,1536) f32
  const float* gw     = (const float*)d_in[3];  // (1,) f32
  float* out = (float*)d_out;                   // [total, reproj]

  float* partials = (float*)d_ws;               // [0,G_RES): residual, [G_RES,G_ALL): prior

  fused_kernel<<<G_ALL, BLK, 0, stream>>>(inputs, hom, rec, partials);
  finalize_kernel<<<1, BLK, 0, stream>>>(partials, G_RES, partials + G_RES, G_PRI, gw, out);
}